// SAGEEncoder_22385369547049
// MI455X (gfx1250) — compile-verified
//
#include <hip/hip_runtime.h>

#define FH 64  // feature width H

typedef __attribute__((ext_vector_type(2))) float v2f;
typedef __attribute__((ext_vector_type(8))) float v8f;

static __device__ __forceinline__ v8f wmma_f32(v2f a, v2f b, v8f c) {
  // V_WMMA_F32_16X16X4_F32 : D(16x16 f32) = A(16x4 f32) * B(4x16 f32) + C
  return __builtin_amdgcn_wmma_f32_16x16x4_f32(
      /*neg_a=*/false, a, /*neg_b=*/false, b,
      /*c_mod=*/(short)0, c, /*reuse_a=*/false, /*reuse_b=*/false);
}

// ---------------------------------------------------------------------------
// Edge scatter: agg[dst] += feat[src], deg[dst] += 1 (deg optional).
// One edge handled by 16 consecutive threads (4 floats each, float4 loads).
// agg (25.6MB) is L2-resident -> atomics resolve in L2.
// ---------------------------------------------------------------------------
__global__ void __launch_bounds__(256)
scatter_kernel(const float* __restrict__ feat,
               const int* __restrict__ src,
               const int* __restrict__ dst,
               float* __restrict__ agg,
               float* __restrict__ deg,   // may be nullptr
               int E) {
  int idx = blockIdx.x * blockDim.x + threadIdx.x;
  int e = idx >> 4;
  if (e >= E) return;
  int c = idx & 15;
  int s = src[e];
  int d = dst[e];
  const float4 v = *reinterpret_cast<const float4*>(feat + (size_t)s * FH + c * 4);
  float* ap = agg + (size_t)d * FH + c * 4;
  atomicAdd(ap + 0, v.x);
  atomicAdd(ap + 1, v.y);
  atomicAdd(ap + 2, v.z);
  atomicAdd(ap + 3, v.w);
  if (deg != nullptr && c == 0) atomicAdd(deg + d, 1.0f);
}

// ---------------------------------------------------------------------------
// out[16x64 tile] = (agg/max(deg,1)) @ Wl^T + x @ Wr^T + bl  (optional ReLU)
// One wave32 per 16-row tile; 4 waves per block; Wl/Wr/bl staged in LDS.
// 16 k-blocks (K=4 each) x 4 col-tiles x 2 matrices = 128 WMMAs per tile.
// ---------------------------------------------------------------------------
__global__ void __launch_bounds__(128)
sage_linear_kernel(const float* __restrict__ agg,
                   const float* __restrict__ xin,
                   const float* __restrict__ deg,
                   const float* __restrict__ Wl,
                   const float* __restrict__ bl,
                   const float* __restrict__ Wr,
                   float* __restrict__ out,
                   int n, int do_relu) {
  __shared__ float sWl[FH * FH];
  __shared__ float sWr[FH * FH];
  __shared__ float sbl[FH];

  const int tid = threadIdx.x;
  for (int i = tid; i < FH * FH; i += 128) { sWl[i] = Wl[i]; sWr[i] = Wr[i]; }
  if (tid < FH) sbl[tid] = bl[tid];
  __syncthreads();

  const int wave = tid >> 5;
  const int lane = tid & 31;
  const int lr   = lane & 15;   // row (A) / col (B) within 16
  const int half = lane >> 4;   // selects k offset pair
  const int tile = blockIdx.x * 4 + wave;
  const int r0   = tile * 16;
  const bool active = (r0 < n);
  const int arow = active ? min(r0 + lr, n - 1) : lr;  // clamped, in-bounds

  const float id = 1.0f / fmaxf(deg[arow], 1.0f);

  v8f acc[4];
  #pragma unroll
  for (int ct = 0; ct < 4; ++ct) {
    const float b = sbl[ct * 16 + lr];   // bias depends on column only
    #pragma unroll
    for (int i = 0; i < 8; ++i) acc[ct][i] = b;
  }

  const float* aggr = agg + (size_t)arow * FH;
  const float* xr   = xin + (size_t)arow * FH;

  #pragma unroll
  for (int kb = 0; kb < 16; ++kb) {
    const int k0 = kb * 4 + half * 2;
    const float2 t = *reinterpret_cast<const float2*>(aggr + k0);
    const float2 u = *reinterpret_cast<const float2*>(xr + k0);
    v2f aM; aM[0] = t.x * id; aM[1] = t.y * id;   // mean row fragment
    v2f aX; aX[0] = u.x;      aX[1] = u.y;        // x row fragment
    #pragma unroll
    for (int ct = 0; ct < 4; ++ct) {
      const int ncol = ct * 16 + lr;
      v2f bL, bR;
      bL[0] = sWl[ncol * FH + k0]; bL[1] = sWl[ncol * FH + k0 + 1];
      bR[0] = sWr[ncol * FH + k0]; bR[1] = sWr[ncol * FH + k0 + 1];
      acc[ct] = wmma_f32(aM, bL, acc[ct]);
      acc[ct] = wmma_f32(aX, bR, acc[ct]);
    }
  }

  if (do_relu) {
    #pragma unroll
    for (int ct = 0; ct < 4; ++ct)
      #pragma unroll
      for (int i = 0; i < 8; ++i) acc[ct][i] = fmaxf(acc[ct][i], 0.0f);
  }

  if (active) {
    #pragma unroll
    for (int ct = 0; ct < 4; ++ct)
      #pragma unroll
      for (int i = 0; i < 8; ++i)
        out[(size_t)(r0 + half * 8 + i) * FH + ct * 16 + lr] = acc[ct][i];
  }
}

// ---------------------------------------------------------------------------
// h = relu(h @ W1^T + b1) @ W2^T + b2   (in-place, row-tile independent)
// Two WMMA GEMMs fused; intermediate tile transposed D-layout -> A-layout
// through per-wave LDS scratch.
// ---------------------------------------------------------------------------
__global__ void __launch_bounds__(128)
mlp_kernel(float* __restrict__ h,
           const float* __restrict__ W1, const float* __restrict__ b1,
           const float* __restrict__ W2, const float* __restrict__ b2,
           int n) {
  __shared__ float sW1[FH * FH];
  __shared__ float sW2[FH * FH];
  __shared__ float sb1[FH];
  __shared__ float sb2[FH];
  __shared__ float tbuf[4][16 * 65];   // +1 pad against bank conflicts

  const int tid = threadIdx.x;
  for (int i = tid; i < FH * FH; i += 128) { sW1[i] = W1[i]; sW2[i] = W2[i]; }
  if (tid < FH) { sb1[tid] = b1[tid]; sb2[tid] = b2[tid]; }
  __syncthreads();

  const int wave = tid >> 5;
  const int lane = tid & 31;
  const int lr   = lane & 15;
  const int half = lane >> 4;
  const int tile = blockIdx.x * 4 + wave;
  const int r0   = tile * 16;
  const bool active = (r0 < n);
  const int arow = active ? min(r0 + lr, n - 1) : lr;

  // ---- stage 1: t = relu(h @ W1^T + b1) ----
  v8f acc[4];
  #pragma unroll
  for (int ct = 0; ct < 4; ++ct) {
    const float b = sb1[ct * 16 + lr];
    #pragma unroll
    for (int i = 0; i < 8; ++i) acc[ct][i] = b;
  }
  const float* hr = h + (size_t)arow * FH;
  #pragma unroll
  for (int kb = 0; kb < 16; ++kb) {
    const int k0 = kb * 4 + half * 2;
    const float2 u = *reinterpret_cast<const float2*>(hr + k0);
    v2f a; a[0] = u.x; a[1] = u.y;
    #pragma unroll
    for (int ct = 0; ct < 4; ++ct) {
      const int ncol = ct * 16 + lr;
      v2f b1f; b1f[0] = sW1[ncol * FH + k0]; b1f[1] = sW1[ncol * FH + k0 + 1];
      acc[ct] = wmma_f32(a, b1f, acc[ct]);
    }
  }
  // ReLU + write tile to LDS in row-major (D layout -> memory order)
  #pragma unroll
  for (int ct = 0; ct < 4; ++ct)
    #pragma unroll
    for (int i = 0; i < 8; ++i)
      tbuf[wave][(half * 8 + i) * 65 + ct * 16 + lr] = fmaxf(acc[ct][i], 0.0f);
  __syncthreads();

  // ---- stage 2: out = t @ W2^T + b2 ----
  v8f acc2[4];
  #pragma unroll
  for (int ct = 0; ct < 4; ++ct) {
    const float b = sb2[ct * 16 + lr];
    #pragma unroll
    for (int i = 0; i < 8; ++i) acc2[ct][i] = b;
  }
  #pragma unroll
  for (int kb = 0; kb < 16; ++kb) {
    const int k0 = kb * 4 + half * 2;
    v2f a; a[0] = tbuf[wave][lr * 65 + k0]; a[1] = tbuf[wave][lr * 65 + k0 + 1];
    #pragma unroll
    for (int ct = 0; ct < 4; ++ct) {
      const int ncol = ct * 16 + lr;
      v2f b2f; b2f[0] = sW2[ncol * FH + k0]; b2f[1] = sW2[ncol * FH + k0 + 1];
      acc2[ct] = wmma_f32(a, b2f, acc2[ct]);
    }
  }

  if (active) {
    #pragma unroll
    for (int ct = 0; ct < 4; ++ct)
      #pragma unroll
      for (int i = 0; i < 8; ++i)
        h[(size_t)(r0 + half * 8 + i) * FH + ct * 16 + lr] = acc2[ct][i];
  }
}

// ---------------------------------------------------------------------------
extern "C" void kernel_launch(void* const* d_in, const int* in_sizes, int n_in,
                              void* d_out, int out_size, void* d_ws, size_t ws_size,
                              hipStream_t stream) {
  const float* x   = (const float*)d_in[0];
  const int*   ei  = (const int*)d_in[1];
  const float* Wl1 = (const float*)d_in[2];
  const float* bl1 = (const float*)d_in[3];
  const float* Wr1 = (const float*)d_in[4];
  const float* W1  = (const float*)d_in[5];
  const float* b1  = (const float*)d_in[6];
  const float* W2  = (const float*)d_in[7];
  const float* b2  = (const float*)d_in[8];
  const float* Wl2 = (const float*)d_in[9];
  const float* bl2 = (const float*)d_in[10];
  const float* Wr2 = (const float*)d_in[11];

  const int n = in_sizes[0] / FH;        // 100000
  const int E = in_sizes[1] / 2;         // 3200000
  const int* src = ei;                   // edge_index[0]
  const int* dst = ei + E;               // edge_index[1]

  const size_t aggBytes = (size_t)n * FH * sizeof(float);
  const size_t degBytes = ((size_t)n * sizeof(float) + 255) & ~(size_t)255;

  char* ws = (char*)d_ws;
  float* agg = (float*)ws;                         // [n, 64]
  float* deg = (float*)(ws + aggBytes);            // [n]
  float* h   = (float*)(ws + aggBytes + degBytes); // [n, 64]

  float* out = (float*)d_out;

  // ---- conv1 aggregation ----
  hipMemsetAsync(agg, 0, aggBytes, stream);
  hipMemsetAsync(deg, 0, (size_t)n * sizeof(float), stream);

  const long scatterThreads = (long)E * 16;
  const int  scatterBlocks  = (int)((scatterThreads + 255) / 256);
  scatter_kernel<<<scatterBlocks, 256, 0, stream>>>(x, src, dst, agg, deg, E);

  const int tiles = (n + 15) / 16;
  const int gemmBlocks = (tiles + 3) / 4;

  // h = relu(sage_conv1(x))
  sage_linear_kernel<<<gemmBlocks, 128, 0, stream>>>(agg, x, deg, Wl1, bl1, Wr1,
                                                     h, n, /*relu=*/1);
  // h = mlp(h)   (in-place, row independent)
  mlp_kernel<<<gemmBlocks, 128, 0, stream>>>(h, W1, b1, W2, b2, n);

  // ---- conv2 aggregation (degree reused, same graph) ----
  hipMemsetAsync(agg, 0, aggBytes, stream);
  scatter_kernel<<<scatterBlocks, 256, 0, stream>>>(h, src, dst, agg, nullptr, E);

  // out = sage_conv2(h)   (no relu)
  sage_linear_kernel<<<gemmBlocks, 128, 0, stream>>>(agg, h, deg, Wl2, bl2, Wr2,
                                                     out, n, /*relu=*/0);
}